// VolumeSDFRenderer_68968584839521
// MI455X (gfx1250) — compile-verified
//
#include <hip/hip_runtime.h>

// ---------------------------------------------------------------------------
// VolumeSDF renderer for MI455X (gfx1250).
// Memory-bound (168 MB @ 23.3 TB/s ~ 7 us floor). No matmul structure (the
// color einsum is a per-ray batched dot, B is not shared), so the CDNA5 path
// that matters is the Tensor Data Mover: double-buffered tensor_load_to_lds
// DMAs tracked with TENSORcnt, overlapping HBM traffic with the wave-level
// transmittance scan (ds-permute shuffles + v_exp_f32).
// ---------------------------------------------------------------------------

typedef unsigned int v4u __attribute__((ext_vector_type(4)));
typedef int          v4i __attribute__((ext_vector_type(4)));
typedef int          v8i __attribute__((ext_vector_type(8)));

#define S_SAMPLES       128
#define RAYS_PER_TILE   8
#define TILES_PER_BLOCK 8
#define BUF_FLOATS      5120   // 1024 sdf + 1024 depth + 3072 color per buffer

// Build a D# (CDNA5 ISA) for a 2-D row-major f32 tile:
//   tile = tile_rows x row_elems, tensor rows remaining = rows_remaining
//   (rows past the tensor edge read back as zero -> safe trailing tiles).
__device__ __forceinline__ void tdm_load_2d(const void* gptr, unsigned lds_off,
                                            unsigned row_elems,
                                            unsigned rows_remaining,
                                            unsigned tile_rows) {
  unsigned long long ga = (unsigned long long)gptr;
  v4u g0;
  g0.x = 1u;                                               // count=1 (valid), user mode
  g0.y = lds_off;                                          // lds_addr (bytes)
  g0.z = (unsigned)ga;                                     // global_addr[31:0]
  g0.w = (unsigned)((ga >> 32) & 0x01FFFFFFull)            // global_addr[56:32]
         | (2u << 30);                                     // type = 2 ("image")
  v8i g1;
  g1[0] = (int)0x00020000u;                                // data_size=2 -> 4 bytes
  g1[1] = (int)((row_elems & 0xFFFFu) << 16);              // tensor_dim0[15:0] @ bits63:48
  g1[2] = (int)(((row_elems >> 16) & 0xFFFFu)              // tensor_dim0[31:16]
                | ((rows_remaining & 0xFFFFu) << 16));     // tensor_dim1[15:0]
  g1[3] = (int)(((rows_remaining >> 16) & 0xFFFFu)         // tensor_dim1[31:16]
                | ((row_elems & 0xFFFFu) << 16));          // tile_dim0
  g1[4] = (int)(tile_rows & 0xFFFFu);                      // tile_dim1 (tile_dim2=0)
  g1[5] = (int)row_elems;                                  // tensor_dim0_stride[31:0]
  g1[6] = 0;                                               // stride[47:32]=0, dim1_stride=0
  g1[7] = 0;
  v4i z4 = {0, 0, 0, 0};
#if __has_include(<hip/amd_detail/amd_gfx1250_TDM.h>)
  // amdgpu-toolchain (clang-23 / therock-10.0): 6-arg form
  v8i z8 = {0, 0, 0, 0, 0, 0, 0, 0};
  __builtin_amdgcn_tensor_load_to_lds(g0, g1, z4, z4, z8, 0);
#else
  // ROCm 7.2 (clang-22): 5-arg form
  __builtin_amdgcn_tensor_load_to_lds(g0, g1, z4, z4, 0);
#endif
}

__device__ __forceinline__ void issue_tile(const float* sdf, const float* depth,
                                           const float* color, int ray0, int N,
                                           unsigned buf_off) {
  const unsigned rem = (unsigned)(N - ray0);
  tdm_load_2d(sdf   + (size_t)ray0 * S_SAMPLES,     buf_off,                 S_SAMPLES,     rem, RAYS_PER_TILE);
  tdm_load_2d(depth + (size_t)ray0 * S_SAMPLES,     buf_off + 1024u * 4u,    S_SAMPLES,     rem, RAYS_PER_TILE);
  tdm_load_2d(color + (size_t)ray0 * S_SAMPLES * 3, buf_off + 2048u * 4u,    S_SAMPLES * 3, rem, RAYS_PER_TILE);
}

__global__ __launch_bounds__(256) void volsdf_render(
    const float* __restrict__ sdf, const float* __restrict__ color,
    const float* __restrict__ depth, float* __restrict__ out, int N) {
  __shared__ float lds[2][BUF_FLOATS];
  const int wv   = (int)(threadIdx.x >> 5);   // wave id, wave32
  const int lane = (int)(threadIdx.x & 31);
  // Flat-LDS low 32 bits are the workgroup-relative LDS byte offset.
  const unsigned lds_base = (unsigned)(unsigned long long)(size_t)&lds[0][0];
  const int tile0 = (int)blockIdx.x * TILES_PER_BLOCK;

  // Prologue: prefetch tile 0 into buffer 0 via TDM (wave 0 issues; TDM
  // ignores EXEC, done signal goes to the issuing wave's TENSORcnt).
  if (wv == 0 && tile0 * RAYS_PER_TILE < N)
    issue_tile(sdf, depth, color, tile0 * RAYS_PER_TILE, N, lds_base);

  for (int t = 0; t < TILES_PER_BLOCK; ++t) {
    const int ray0 = (tile0 + t) * RAYS_PER_TILE;
    if (ray0 >= N) break;                     // block-uniform
    const int b = t & 1;
    if (wv == 0) {
      const int nray0 = ray0 + RAYS_PER_TILE;
      if (t + 1 < TILES_PER_BLOCK && nray0 < N) {
        issue_tile(sdf, depth, color, nray0,
                   N, lds_base + (unsigned)((b ^ 1) * BUF_FLOATS * 4));
        // 3 loads of the *next* tile stay in flight; current tile's 3 done.
        __builtin_amdgcn_s_wait_tensorcnt(3);
      } else {
        __builtin_amdgcn_s_wait_tensorcnt(0);
      }
    }
    __syncthreads();                          // publish LDS tile to all waves

    const float* sdf_s = &lds[b][wv * S_SAMPLES];
    const float* dep_s = &lds[b][1024 + wv * S_SAMPLES];
    const float* col_s = &lds[b][2048 + wv * S_SAMPLES * 3];

    float acc0 = 0.f, acc1 = 0.f, acc2 = 0.f;
    float carry = 0.f;                        // running cumsum across 32-chunks
#pragma unroll
    for (int j = 0; j < 4; ++j) {
      const int s = lane + j * 32;
      // density = ALPHA * LaplaceCDF(-sdf, BETA); 1/BETA = 20, ALPHA = 10
      const float sd = sdf_s[s];
      const float e  = __expf(-fabsf(sd) * 20.0f);
      const float dens = (sd > 0.f) ? (5.0f * e) : (10.0f - 5.0f * e);
      // delta: next sample resides in the LDS depth tile
      const float d0 = dep_s[s];
      float delta;
      if (s < S_SAMPLES - 1) delta = dep_s[s + 1] - d0;
      else                   delta = 1.0e10f;  // FAR_DELTA
      const float tau = dens * delta;
      // inclusive wave scan of tau (Hillis-Steele over 32 lanes)
      float x = tau;
#pragma unroll
      for (int off = 1; off < 32; off <<= 1) {
        const float y = __shfl_up(x, off, 32);
        if (lane >= off) x += y;
      }
      const float Tr = __expf(-(carry + x));                  // exp(-inclusive cumsum)
      const float w  = Tr * (1.0f - __expf(1.0e-10f - tau));  // T * (1 - exp(-tau+eps))
      acc0 += w * col_s[3 * s + 0];
      acc1 += w * col_s[3 * s + 1];
      acc2 += w * col_s[3 * s + 2];
      carry += __shfl(x, 31, 32);             // chunk total -> next chunk offset
    }
    // cross-lane RGB reduction
#pragma unroll
    for (int off = 16; off > 0; off >>= 1) {
      acc0 += __shfl_down(acc0, off, 32);
      acc1 += __shfl_down(acc1, off, 32);
      acc2 += __shfl_down(acc2, off, 32);
    }
    const int ray = ray0 + wv;
    if (lane == 0 && ray < N) {
      out[3 * ray + 0] = acc0;
      out[3 * ray + 1] = acc1;
      out[3 * ray + 2] = acc2;
    }
    __syncthreads();                          // compute done before buf reuse
  }
}

extern "C" void kernel_launch(void* const* d_in, const int* in_sizes, int n_in,
                              void* d_out, int out_size, void* d_ws, size_t ws_size,
                              hipStream_t stream) {
  (void)n_in; (void)out_size; (void)d_ws; (void)ws_size;
  const float* sdf   = (const float*)d_in[0];  // [N,128]
  const float* color = (const float*)d_in[1];  // [N,128,3]
  const float* depth = (const float*)d_in[2];  // [N,128]
  float* out = (float*)d_out;                  // [N,3]
  const int N = in_sizes[0] / S_SAMPLES;
  const int rays_per_block = RAYS_PER_TILE * TILES_PER_BLOCK;  // 64
  const int grid = (N + rays_per_block - 1) / rays_per_block;
  volsdf_render<<<grid, 256, 0, stream>>>(sdf, color, depth, out, N);
}